// Model_22376779612219
// MI455X (gfx1250) — compile-verified
//
#include <hip/hip_runtime.h>
#include <math.h>

#define BB 2
#define SEQ 512
#define NVAR 32
#define DM 512
#define DST 16
#define DCONV 4
#define DIN 1024
#define DTR 32
#define PN 64
#define PL 16
#define STRIDE_ 8
#define BN 64          /* BB*NVAR */
#define TOK 4096       /* BN*PN */
#define PRED 96
#define NF 32768
#define EPSF 1e-5f

typedef float v8f __attribute__((ext_vector_type(8)));
typedef float v2f __attribute__((ext_vector_type(2)));

// ---------------- per-(b,var) mean / std over SEQ ----------------
__global__ void stats_kernel(const float* __restrict__ x, float* __restrict__ mean,
                             float* __restrict__ stdv) {
  int bn = blockIdx.x; int b = bn / NVAR, v = bn % NVAR;
  const float* xp = x + (size_t)b * SEQ * NVAR + v;
  float s = 0.f, s2 = 0.f;
  for (int t = threadIdx.x; t < SEQ; t += blockDim.x) {
    float u = xp[(size_t)t * NVAR]; s += u; s2 += u * u;
  }
  __shared__ float r1[256], r2[256];
  r1[threadIdx.x] = s; r2[threadIdx.x] = s2; __syncthreads();
  for (int o = 128; o > 0; o >>= 1) {
    if (threadIdx.x < o) { r1[threadIdx.x] += r1[threadIdx.x + o]; r2[threadIdx.x] += r2[threadIdx.x + o]; }
    __syncthreads();
  }
  if (threadIdx.x == 0) {
    float m = r1[0] / SEQ;
    float var = r2[0] / SEQ - m * m;
    mean[bn] = m; stdv[bn] = sqrtf(var + EPSF);
  }
}

// ---------------- patch embed + sinusoidal PE ----------------
__global__ void embed_kernel(const float* __restrict__ x, const float* __restrict__ mean,
                             const float* __restrict__ stdv, const float* __restrict__ val_emb,
                             float* __restrict__ enc) {
  int blk = blockIdx.x; int bn = blk / PN; int p = blk % PN;
  int b = bn / NVAR, v = bn % NVAR;
  __shared__ float sp[PL];
  if (threadIdx.x < PL) {
    int i = p * STRIDE_ + threadIdx.x; if (i >= SEQ) i = SEQ - 1;   // edge pad
    sp[threadIdx.x] = (x[(size_t)b * SEQ * NVAR + (size_t)i * NVAR + v] - mean[bn]) / stdv[bn];
  }
  __syncthreads();
  const float LOGC = logf(10000.0f) / (float)DM;
  for (int d = threadIdx.x; d < DM; d += blockDim.x) {
    float freq = expf(-(float)(d & ~1) * LOGC);
    float ang = (float)p * freq;
    float acc = (d & 1) ? cosf(ang) : sinf(ang);
#pragma unroll
    for (int k = 0; k < PL; k++) acc += sp[k] * val_emb[k * DM + d];
    enc[(size_t)bn * PN * DM + (size_t)p * DM + d] = acc;
  }
}

// ---------------- generic fp32 WMMA GEMM: C[b] = A[b] @ B[b] (+bias)(+act) ----------------
// one wave per block; wave computes a 32x32 tile (2x2 fragments, 4 accumulators).
// Per K-step: 2 x b64 A loads + 4 x b32 B loads feed 4 v_wmma_f32_16x16x4_f32.
// act: 0=none, 1=softplus
__global__ void gemm_wmma(const float* __restrict__ A, const float* __restrict__ Bm,
                          float* __restrict__ C, int K,
                          int lda, int ldb, int ldc,
                          long Abs, long Bbs, long Cbs,
                          const float* __restrict__ bias, int accum, int act) {
  int lane = threadIdx.x;
  int n0 = blockIdx.x * 32;
  int m0 = blockIdx.y * 32;
  const float* Ab = A + (long)blockIdx.z * Abs;
  const float* Bb = Bm + (long)blockIdx.z * Bbs;
  float* Cb = C + (long)blockIdx.z * Cbs;
  int r16 = lane & 15;
  int kh = (lane >> 4) * 2;
  int rbase = (lane >> 4) * 8;
  v8f c00 = {0.f, 0.f, 0.f, 0.f, 0.f, 0.f, 0.f, 0.f};
  v8f c01 = c00, c10 = c00, c11 = c00;
  if (accum) {
#pragma unroll
    for (int r = 0; r < 8; r++) {
      size_t ro0 = (size_t)(m0 + rbase + r) * ldc;
      size_t ro1 = (size_t)(m0 + 16 + rbase + r) * ldc;
      c00[r] = Cb[ro0 + n0 + r16];
      c01[r] = Cb[ro0 + n0 + 16 + r16];
      c10[r] = Cb[ro1 + n0 + r16];
      c11[r] = Cb[ro1 + n0 + 16 + r16];
    }
  }
  const float* arow0 = Ab + (size_t)(m0 + r16) * lda + kh;
  const float* arow1 = arow0 + (size_t)16 * lda;
  for (int kk = 0; kk < K; kk += 4) {
    v2f a0, a1, b0, b1;
    a0[0] = arow0[kk];
    a0[1] = arow0[kk + 1];
    a1[0] = arow1[kk];
    a1[1] = arow1[kk + 1];
    const float* brow = Bb + (size_t)(kk + kh) * ldb;
    b0[0] = brow[n0 + r16];
    b1[0] = brow[n0 + 16 + r16];
    b0[1] = brow[(size_t)ldb + n0 + r16];
    b1[1] = brow[(size_t)ldb + n0 + 16 + r16];
    c00 = __builtin_amdgcn_wmma_f32_16x16x4_f32(false, a0, false, b0, (short)0, c00, false, false);
    c01 = __builtin_amdgcn_wmma_f32_16x16x4_f32(false, a0, false, b1, (short)0, c01, false, false);
    c10 = __builtin_amdgcn_wmma_f32_16x16x4_f32(false, a1, false, b0, (short)0, c10, false, false);
    c11 = __builtin_amdgcn_wmma_f32_16x16x4_f32(false, a1, false, b1, (short)0, c11, false, false);
  }
  float bv0 = 0.f, bv1 = 0.f;
  if (bias) { bv0 = bias[n0 + r16]; bv1 = bias[n0 + 16 + r16]; }
#pragma unroll
  for (int r = 0; r < 8; r++) {
    size_t ro0 = (size_t)(m0 + rbase + r) * ldc;
    size_t ro1 = (size_t)(m0 + 16 + rbase + r) * ldc;
    float v00 = c00[r] + bv0, v01 = c01[r] + bv1;
    float v10 = c10[r] + bv0, v11 = c11[r] + bv1;
    if (act == 1) {
      v00 = (v00 > 20.f) ? v00 : log1pf(expf(v00));
      v01 = (v01 > 20.f) ? v01 : log1pf(expf(v01));
      v10 = (v10 > 20.f) ? v10 : log1pf(expf(v10));
      v11 = (v11 > 20.f) ? v11 : log1pf(expf(v11));
    }
    Cb[ro0 + n0 + r16] = v00;
    Cb[ro0 + n0 + 16 + r16] = v01;
    Cb[ro1 + n0 + r16] = v10;
    Cb[ro1 + n0 + 16 + r16] = v11;
  }
}

// ---------------- causal depthwise conv (k=4) + bias + SiLU ----------------
__global__ void conv_silu_kernel(const float* __restrict__ xxres, const float* __restrict__ conv_w,
                                 const float* __restrict__ conv_b, float* __restrict__ xxs) {
  long idx = (long)blockIdx.x * blockDim.x + threadIdx.x;
  if (idx >= (long)BN * PN * DIN) return;
  int d = (int)(idx % DIN);
  long t = idx / DIN;
  int l = (int)(t % PN);
  int bn = (int)(t / PN);
  const float* xb = xxres + (size_t)bn * PN * 2 * DIN + d;
  float acc = conv_b[d];
#pragma unroll
  for (int j = 0; j < DCONV; j++) {
    int ls = l + j - (DCONV - 1);
    if (ls >= 0) acc += conv_w[d * DCONV + j] * xb[(size_t)ls * 2 * DIN];
  }
  acc = acc / (1.f + __expf(-acc));
  xxs[idx] = acc;
}

// ---------------- selective scan; writes y (in place of xxs) ----------------
__global__ void scan_kernel(const float* __restrict__ delta, const float* __restrict__ xpo,
                            const float* __restrict__ xxres, const float* __restrict__ A_log,
                            const float* __restrict__ Dp, float* __restrict__ xxs) {
  int bn = blockIdx.x >> 2;
  int d = ((blockIdx.x & 3) << 8) + threadIdx.x;
  float a[DST], h[DST];
#pragma unroll
  for (int n = 0; n < DST; n++) { a[n] = -__expf(A_log[d * DST + n]); h[n] = 0.f; }
  float Dv = Dp[d];
  __shared__ float sB[DST], sC[DST];
  const float* xpob = xpo + (size_t)bn * PN * 64;
  for (int l = 0; l < PN; l++) {
    __syncthreads();
    if (threadIdx.x < DST) sB[threadIdx.x] = xpob[l * 64 + DTR + threadIdx.x];
    else if (threadIdx.x < 2 * DST) sC[threadIdx.x - DST] = xpob[l * 64 + DTR + threadIdx.x];
    __syncthreads();
    size_t off = (size_t)bn * PN * DIN + (size_t)l * DIN + d;
    float dt = delta[off];
    float xv = xxs[off];
    float du = dt * xv;
    float y = 0.f;
#pragma unroll
    for (int n = 0; n < DST; n++) {
      h[n] = __expf(dt * a[n]) * h[n] + du * sB[n];
      y += h[n] * sC[n];
    }
    y += xv * Dv;
    float res = xxres[(size_t)bn * PN * 2 * DIN + (size_t)l * 2 * DIN + DIN + d];
    y *= res / (1.f + __expf(-res));   // * silu(res)
    xxs[off] = y;
  }
}

// ---------------- circulant kernel matrix from filt (irfft) ----------------
__global__ void kmat_kernel(const float* __restrict__ filt_r, const float* __restrict__ filt_i,
                            float* __restrict__ kmat) {
  int n = threadIdx.x;  // 64 threads
  __shared__ float kv[PN];
  float acc = filt_r[0] + filt_r[32] * ((n & 1) ? -1.f : 1.f);
  const float W = 2.f * 3.14159265358979323846f / (float)PN;
  for (int f = 1; f < 32; f++) {
    float th = W * (float)(f * n);
    acc += 2.f * (filt_r[f] * cosf(th) - filt_i[f] * sinf(th));
  }
  kv[n] = acc / (float)PN;
  __syncthreads();
  for (int m = 0; m < PN; m++) kmat[n * PN + m] = kv[(n - m) & 63];
}

// ---------------- layernorm over last dim (512), one row per block ----------------
__global__ void ln_kernel(const float* __restrict__ in, const float* __restrict__ g,
                          const float* __restrict__ b, float* __restrict__ out) {
  size_t o = (size_t)blockIdx.x * DM;
  float v0 = in[o + threadIdx.x], v1 = in[o + threadIdx.x + 256];
  __shared__ float r1[256], r2[256];
  r1[threadIdx.x] = v0 + v1; r2[threadIdx.x] = v0 * v0 + v1 * v1; __syncthreads();
  for (int oo = 128; oo > 0; oo >>= 1) {
    if (threadIdx.x < oo) { r1[threadIdx.x] += r1[threadIdx.x + oo]; r2[threadIdx.x] += r2[threadIdx.x + oo]; }
    __syncthreads();
  }
  __shared__ float sm, sr;
  if (threadIdx.x == 0) {
    float m = r1[0] / DM; float var = r2[0] / DM - m * m;
    sm = m; sr = rsqrtf(var + EPSF);
  }
  __syncthreads();
  out[o + threadIdx.x]       = (v0 - sm) * sr * g[threadIdx.x]       + b[threadIdx.x];
  out[o + threadIdx.x + 256] = (v1 - sm) * sr * g[threadIdx.x + 256] + b[threadIdx.x + 256];
}

// ---------------- gated combine + layernorm, writes enc in place ----------------
__global__ void combine_ln_kernel(const float* __restrict__ gpre, const float* __restrict__ ht,
                                  const float* __restrict__ hf, const float* __restrict__ g,
                                  const float* __restrict__ b, float* __restrict__ enc) {
  size_t o = (size_t)blockIdx.x * DM;
  float vv[2];
#pragma unroll
  for (int i = 0; i < 2; i++) {
    int d = threadIdx.x + i * 256;
    float gg = 1.f / (1.f + __expf(-gpre[o + d]));
    vv[i] = gg * ht[o + d] + (1.f - gg) * hf[o + d] + enc[o + d];
  }
  __shared__ float r1[256], r2[256];
  r1[threadIdx.x] = vv[0] + vv[1];
  r2[threadIdx.x] = vv[0] * vv[0] + vv[1] * vv[1];
  __syncthreads();
  for (int oo = 128; oo > 0; oo >>= 1) {
    if (threadIdx.x < oo) { r1[threadIdx.x] += r1[threadIdx.x + oo]; r2[threadIdx.x] += r2[threadIdx.x + oo]; }
    __syncthreads();
  }
  __shared__ float sm, sr;
  if (threadIdx.x == 0) {
    float m = r1[0] / DM; float var = r2[0] / DM - m * m;
    sm = m; sr = rsqrtf(var + EPSF);
  }
  __syncthreads();
#pragma unroll
  for (int i = 0; i < 2; i++) {
    int d = threadIdx.x + i * 256;
    enc[o + d] = (vv[i] - sm) * sr * g[d] + b[d];
  }
}

// ---------------- transpose enc (bn,p,d) -> encT (bn, d*64+p) for head ----------------
__global__ void transpose_kernel(const float* __restrict__ enc, float* __restrict__ encT) {
  long idx = (long)blockIdx.x * blockDim.x + threadIdx.x;
  if (idx >= (long)BN * PN * DM) return;
  int d = (int)(idx % DM);
  long t = idx / DM;
  int p = (int)(t % PN);
  int bn = (int)(t / PN);
  encT[(size_t)bn * NF + (size_t)d * PN + p] = enc[idx];
}

// ---------------- de-normalize + transpose to (B, PRED, NV) ----------------
__global__ void out_kernel(const float* __restrict__ headout, const float* __restrict__ mean,
                           const float* __restrict__ stdv, float* __restrict__ out) {
  int idx = blockIdx.x * blockDim.x + threadIdx.x;
  if (idx >= BB * PRED * NVAR) return;
  int v = idx % NVAR;
  int t = (idx / NVAR) % PRED;
  int b = idx / (NVAR * PRED);
  int bn = b * NVAR + v;
  out[idx] = headout[(size_t)bn * PRED + t] * stdv[bn] + mean[bn];
}

struct LayerP {
  const float *in_proj, *conv_w, *conv_b, *x_proj, *dt_w, *dt_b, *A_log, *D, *out_proj,
              *filt_r, *filt_i, *fproj_w, *fproj_b, *fln_g, *fln_b, *gate_w, *gate_b, *ln_g, *ln_b;
};

extern "C" void kernel_launch(void* const* d_in, const int* in_sizes, int n_in,
                              void* d_out, int out_size, void* d_ws, size_t ws_size,
                              hipStream_t stream) {
  (void)out_size; (void)ws_size;
  const float* x_enc = (const float*)d_in[0];
  LayerP L[2];
  const float *val_emb = nullptr, *head_w = nullptr, *head_b = nullptr;
  int p = 4;
  bool sorted_order = (n_in > 4 && in_sizes[4] == PRED);  // head_b first => sorted pytree keys
  if (sorted_order) {
    head_b = (const float*)d_in[p++];
    head_w = (const float*)d_in[p++];
    for (int li = 0; li < 2; li++) {
      L[li].A_log   = (const float*)d_in[p++];
      L[li].D       = (const float*)d_in[p++];
      L[li].conv_b  = (const float*)d_in[p++];
      L[li].conv_w  = (const float*)d_in[p++];
      L[li].dt_b    = (const float*)d_in[p++];
      L[li].dt_w    = (const float*)d_in[p++];
      L[li].filt_i  = (const float*)d_in[p++];
      L[li].filt_r  = (const float*)d_in[p++];
      L[li].fln_b   = (const float*)d_in[p++];
      L[li].fln_g   = (const float*)d_in[p++];
      L[li].fproj_b = (const float*)d_in[p++];
      L[li].fproj_w = (const float*)d_in[p++];
      L[li].gate_b  = (const float*)d_in[p++];
      L[li].gate_w  = (const float*)d_in[p++];
      L[li].in_proj = (const float*)d_in[p++];
      L[li].ln_b    = (const float*)d_in[p++];
      L[li].ln_g    = (const float*)d_in[p++];
      L[li].out_proj= (const float*)d_in[p++];
      L[li].x_proj  = (const float*)d_in[p++];
    }
    val_emb = (const float*)d_in[p++];
  } else {
    val_emb = (const float*)d_in[p++];
    for (int li = 0; li < 2; li++) {
      L[li].in_proj = (const float*)d_in[p++];
      L[li].conv_w  = (const float*)d_in[p++];
      L[li].conv_b  = (const float*)d_in[p++];
      L[li].x_proj  = (const float*)d_in[p++];
      L[li].dt_w    = (const float*)d_in[p++];
      L[li].dt_b    = (const float*)d_in[p++];
      L[li].A_log   = (const float*)d_in[p++];
      L[li].D       = (const float*)d_in[p++];
      L[li].out_proj= (const float*)d_in[p++];
      L[li].filt_r  = (const float*)d_in[p++];
      L[li].filt_i  = (const float*)d_in[p++];
      L[li].fproj_w = (const float*)d_in[p++];
      L[li].fproj_b = (const float*)d_in[p++];
      L[li].fln_g   = (const float*)d_in[p++];
      L[li].fln_b   = (const float*)d_in[p++];
      L[li].gate_w  = (const float*)d_in[p++];
      L[li].gate_b  = (const float*)d_in[p++];
      L[li].ln_g    = (const float*)d_in[p++];
      L[li].ln_b    = (const float*)d_in[p++];
    }
    head_w = (const float*)d_in[p++];
    head_b = (const float*)d_in[p++];
  }

  // workspace layout (floats)
  float* W = (float*)d_ws;
  float* mean    = W + 0;          // 64
  float* stdv    = W + 64;         // 64
  float* kmat    = W + 192;        // 4096
  float* headout = W + 4288;       // 6144
  float* enc     = W + 16384;      // 2,097,152  (BN,PN,DM)
  float* xxres   = W + 2113536;    // 8,388,608  (BN,PN,2*DIN)
  float* xxs     = W + 10502144;   // 4,194,304  (BN,PN,DIN)  -> ys in place
  float* xpo     = W + 14696448;   //   262,144  (BN,PN,64)
  float* delta   = W + 14958592;   // 4,194,304  (BN,PN,DIN)
  float* ht      = W + 19152896;   // 2,097,152
  float* xfr     = W + 21250048;   // 2,097,152
  float* hf      = W + 23347200;   // 2,097,152
  float* gpre  = xxs;    // reuse (dead after out_proj)
  float* hfpre = delta;  // reuse (dead after scan)
  float* encT  = xxres;  // reuse (dead after layers)

  // 0) normalize stats + patch embed
  stats_kernel<<<BN, 256, 0, stream>>>(x_enc, mean, stdv);
  embed_kernel<<<BN * PN, 256, 0, stream>>>(x_enc, mean, stdv, val_emb, enc);

  for (int li = 0; li < 2; li++) {
    // 1) in_proj: (4096x512)@(512x2048)
    gemm_wmma<<<dim3(2 * DIN / 32, TOK / 32, 1), 32, 0, stream>>>(
        enc, L[li].in_proj, xxres, DM, DM, 2 * DIN, 2 * DIN, 0, 0, 0, nullptr, 0, 0);
    // 2) depthwise conv + SiLU
    conv_silu_kernel<<<(BN * PN * DIN) / 256, 256, 0, stream>>>(xxres, L[li].conv_w, L[li].conv_b, xxs);
    // 3) x_proj: (4096x1024)@(1024x64)
    gemm_wmma<<<dim3(64 / 32, TOK / 32, 1), 32, 0, stream>>>(
        xxs, L[li].x_proj, xpo, DIN, DIN, 64, 64, 0, 0, 0, nullptr, 0, 0);
    // 4) delta = softplus(xpo[:, :32] @ dt_w + dt_b)
    gemm_wmma<<<dim3(DIN / 32, TOK / 32, 1), 32, 0, stream>>>(
        xpo, L[li].dt_w, delta, DTR, 64, DIN, DIN, 0, 0, 0, L[li].dt_b, 0, 1);
    // 5) selective scan (ys written in place of xxs)
    scan_kernel<<<BN * 4, 256, 0, stream>>>(delta, xpo, xxres, L[li].A_log, L[li].D, xxs);
    // 6) ht = ys @ out_proj
    gemm_wmma<<<dim3(DM / 32, TOK / 32, 1), 32, 0, stream>>>(
        xxs, L[li].out_proj, ht, DIN, DIN, DM, DM, 0, 0, 0, nullptr, 0, 0);
    // 7) circulant kernel from filt, then batched circulant matmul = irfft(rfft(x)*filt)
    kmat_kernel<<<1, PN, 0, stream>>>(L[li].filt_r, L[li].filt_i, kmat);
    gemm_wmma<<<dim3(DM / 32, PN / 32, BN), 32, 0, stream>>>(
        kmat, enc, xfr, PN, PN, DM, DM, 0, (long)PN * DM, (long)PN * DM, nullptr, 0, 0);
    // 8) fproj + bias, then layernorm -> hf
    gemm_wmma<<<dim3(DM / 32, TOK / 32, 1), 32, 0, stream>>>(
        xfr, L[li].fproj_w, hfpre, DM, DM, DM, DM, 0, 0, 0, L[li].fproj_b, 0, 0);
    ln_kernel<<<TOK, 256, 0, stream>>>(hfpre, L[li].fln_g, L[li].fln_b, hf);
    // 9) gate pre-activation: gpre = ht@Gw_top + b ; gpre += hf@Gw_bot
    gemm_wmma<<<dim3(DM / 32, TOK / 32, 1), 32, 0, stream>>>(
        ht, L[li].gate_w, gpre, DM, DM, DM, DM, 0, 0, 0, L[li].gate_b, 0, 0);
    gemm_wmma<<<dim3(DM / 32, TOK / 32, 1), 32, 0, stream>>>(
        hf, L[li].gate_w + (size_t)DM * DM, gpre, DM, DM, DM, DM, 0, 0, 0, nullptr, 1, 0);
    // 10) combine + layernorm -> enc
    combine_ln_kernel<<<TOK, 256, 0, stream>>>(gpre, ht, hf, L[li].ln_g, L[li].ln_b, enc);
  }

  // head: encT (64 x 32768) @ head_w (32768 x 96) + head_b
  transpose_kernel<<<(BN * PN * DM) / 256, 256, 0, stream>>>(enc, encT);
  gemm_wmma<<<dim3(PRED / 32, BN / 32, 1), 32, 0, stream>>>(
      encT, head_w, headout, NF, NF, PRED, PRED, 0, 0, 0, head_b, 0, 0);
  out_kernel<<<(BB * PRED * NVAR + 255) / 256, 256, 0, stream>>>(headout, mean, stdv, (float*)d_out);
}